// SPConv2d_23708219474629
// MI455X (gfx1250) — compile-verified
//
#include <hip/hip_runtime.h>

typedef __attribute__((ext_vector_type(2))) float v2f;
typedef __attribute__((ext_vector_type(4))) float v4f;
typedef __attribute__((ext_vector_type(8))) float v8f;

#define Hdim 256
#define Wdim 256
#define Cdim 64
#define Odim 64
#define TILE 16
#define CSTR 68    // padded channel-stride (floats) for LDS [pix][c] arrays:
                   // b64 B-fragment loads are bank-conflict-free (4*l vs 4*l+2)
                   // and rows stay 16B aligned (68*4 = 272 = 16*17)

__global__ __launch_bounds__(256) void spconv2d_wmma_kernel(
    const float* __restrict__ x, const float* __restrict__ core,
    const float* __restrict__ periphery, const float* __restrict__ threshold,
    float* __restrict__ out)
{
  __shared__ float s_v[256 * CSTR];      // [pix][c]: agg, then selected v
  __shared__ float s_ctr[256 * CSTR];    // [pix][c]: center values
  __shared__ float s_tile[18 * 20];      // one channel's halo tile (padded rows)

  const int tid = threadIdx.x;           // 0..255, == pixel index in tile
  const int b  = blockIdx.z;
  const int h0 = blockIdx.y * TILE;
  const int w0 = blockIdx.x * TILE;
  const int ph = tid >> 4;
  const int pw = tid & 15;

  float per[8];
  #pragma unroll
  for (int k = 0; k < 8; ++k) per[k] = periphery[k];
  const float thr = threshold[0];
  // scale (>0) is mathematically irrelevant for the hard mask:
  // sigmoid((div-thr)*scale) > 0.5  <=>  div > thr

  float divacc = 0.f;

  // ---------------- Phase 1: per-channel tap math via LDS halo tile ---------
  for (int c = 0; c < Cdim; ++c) {
    const float* xc = x + ((size_t)(b * Cdim + c)) * (Hdim * Wdim);
    for (int idx = tid; idx < 18 * 18; idx += 256) {
      const int row = idx / 18, col = idx - row * 18;
      const int hy = h0 + row - 1, wx = w0 + col - 1;
      float v = 0.f;
      if (hy >= 0 && hy < Hdim && wx >= 0 && wx < Wdim) {
        v = xc[hy * Wdim + wx];
        if (c + 1 < Cdim)
          __builtin_prefetch(xc + Hdim * Wdim + hy * Wdim + wx, 0, 1);
      }
      s_tile[row * 20 + col] = v;
    }
    __syncthreads();

    const float ctr = s_tile[(ph + 1) * 20 + (pw + 1)];
    float divc = 0.f, aggc = 0.f;
    #pragma unroll
    for (int di = 0; di < 3; ++di) {
      #pragma unroll
      for (int dj = 0; dj < 3; ++dj) {
        const float t = s_tile[(ph + di) * 20 + (pw + dj)];
        const float d = t - ctr;
        divc += d * d;
        const int tap = di * 3 + dj;
        if (tap != 4) aggc += per[tap < 4 ? tap : tap - 1] * t;
      }
    }
    divacc += divc;
    s_v[tid * CSTR + c]   = aggc;
    s_ctr[tid * CSTR + c] = ctr;
    __syncthreads();
  }

  // ---------------- Phase 2: hard mask + in-place select of v (b128 ops) ----
  const bool edge = divacc > thr;
  {
    v4f* vrow = (v4f*)&s_v[tid * CSTR];        // 16B-aligned (272 = 16*17)
    const v4f* crow = (const v4f*)&s_ctr[tid * CSTR];
    #pragma unroll
    for (int c4 = 0; c4 < Cdim / 4; ++c4) {
      v4f vv = edge ? vrow[c4] : crow[c4];
      vrow[c4] = vv;
    }
  }
  __syncthreads();

  // ---------------- Phase 3: out_tile[64x256] = core[64x64] * v[64x256] ----
  // V_WMMA_F32_16X16X4_F32, 16 K-steps of 4. 8 waves: mt = wave&3 (O rows),
  // each covers 8 N-tiles of 16 pixels (one pixel row each).
  const int lane = tid & 31;
  const int wave = tid >> 5;
  const int l16  = lane & 15;
  const int half = lane >> 4;             // 0: lanes 0-15, 1: lanes 16-31
  const int mt   = wave & 3;
  const int ntb  = (wave >> 2) * 8;

  // A fragments: 16x4 f32 tile => lane holds (K = 2*half + vgpr) of row m
  v2f a[16];
  #pragma unroll
  for (int ks = 0; ks < 16; ++ks) {
    const int m = mt * 16 + l16;
    const int k = ks * 4 + half * 2;
    a[ks].x = core[m * Cdim + k];
    a[ks].y = core[m * Cdim + k + 1];
  }

  for (int nti = 0; nti < 8; ++nti) {
    const int nt = ntb + nti;
    const float* vrow = &s_v[(nt * 16 + l16) * CSTR];  // this lane's pixel row
    v8f acc = {};
    #pragma unroll
    for (int ks = 0; ks < 16; ++ks) {
      const int k = ks * 4 + half * 2;
      v2f bfrag = *(const v2f*)&vrow[k];   // single ds_load_b64, conflict-free
      acc = __builtin_amdgcn_wmma_f32_16x16x4_f32(
          /*neg_a=*/false, a[ks], /*neg_b=*/false, bfrag,
          /*c_mod=*/(short)0, acc, /*reuse_a=*/false, /*reuse_b=*/false);
    }
    // D 16x16 f32: VGPR r -> M = r + 8*half, N = l16
    const int hh = h0 + nt;
    #pragma unroll
    for (int r = 0; r < 8; ++r) {
      const int o = mt * 16 + r + 8 * half;
      out[(((size_t)(b * Odim + o)) * Hdim + hh) * Wdim + (w0 + l16)] = acc[r];
    }
  }
}

extern "C" void kernel_launch(void* const* d_in, const int* in_sizes, int n_in,
                              void* d_out, int out_size, void* d_ws, size_t ws_size,
                              hipStream_t stream) {
  const float* x         = (const float*)d_in[0];
  const float* core      = (const float*)d_in[1];
  const float* periphery = (const float*)d_in[2];
  const float* threshold = (const float*)d_in[3];
  // d_in[4] (scale) is provably irrelevant for the hard mask (scale > 0).
  float* out = (float*)d_out;

  dim3 grid(Wdim / TILE, Hdim / TILE, 4);
  spconv2d_wmma_kernel<<<grid, 256, 0, stream>>>(x, core, periphery, threshold, out);
}